// AggrSum_6846177869845
// MI455X (gfx1250) — compile-verified
//
#include <hip/hip_runtime.h>
#include <stdint.h>

// Segment-sum: out[v,:] = sum over n with X[n]==v of H[n,:]
// H: 1.6M x 128 f32 (819 MB, streamed), X: 1.6M int64, out: 100K x 128 f32 (51 MB, L2-resident).
// CDNA5 path: async global->LDS double-buffered staging (ASYNCcnt, fire-and-forget
// DMA that burns no VGPRs/LOADcnt), ds_load_b128 readback, v_readlane broadcast,
// hardware global_atomic_add_f32 scatter absorbed by the 192MB L2.

typedef float v4f __attribute__((ext_vector_type(4)));

#define COLS 128
#define TROWS 16                               // rows per tile per wave
#define ROW_BYTES (COLS * 4)                   // 512
#define TILE_BYTES (TROWS * ROW_BYTES)         // 8192
#define WAVE_LDS_BYTES (2 * TILE_BYTES)        // 16384 (double buffer)
#define WAVES_PER_BLOCK 8
#define BLOCK_LDS_BYTES (WAVES_PER_BLOCK * WAVE_LDS_BYTES)  // 131072

// Issue one tile (exactly TROWS async ops, rows clamped in-bounds). One
// global_load_async_to_lds_b128 moves 16B/lane * 32 lanes = 512B = one row.
// Always issuing exactly TROWS ops keeps the s_wait_asynccnt 0x10 contract
// valid (async loads retire in order).
__device__ __forceinline__ void stage_tile_async(const float* H, int rows, int base,
                                                 uint32_t lds_byte_base, int lane) {
    uint32_t l = lds_byte_base + (uint32_t)(lane * 16);
#pragma unroll
    for (int r = 0; r < TROWS; ++r) {
        int row = base + r;
        if (row >= rows) row = rows - 1;       // clamp: padded rows are never consumed
        uint64_t g = (uint64_t)(uintptr_t)(H + (size_t)row * COLS) + (uint32_t)(lane * 16);
        asm volatile("global_load_async_to_lds_b128 %0, %1, off"
                     :: "v"(l), "v"(g)
                     : "memory");
        l += ROW_BYTES;
    }
}

__global__ __launch_bounds__(256) void seg_scatter(const float* __restrict__ H,
                                                   const long long* __restrict__ X,
                                                   float* __restrict__ out,
                                                   int rows) {
    extern __shared__ v4f smem4[];             // dynamic LDS, 16B aligned
    char* smem = (char*)smem4;

    const int lane  = threadIdx.x & 31;        // wave32
    const int wib   = threadIdx.x >> 5;        // wave index in block
    const int gwave = (blockIdx.x * blockDim.x + threadIdx.x) >> 5;
    const int nwave = (gridDim.x * blockDim.x) >> 5;

    // Workgroup-relative LDS byte offset of this wave's region (generic shared
    // pointer low 32 bits == LDS offset per the flat-aperture mapping).
    const uint32_t wave_lds = (uint32_t)(uintptr_t)smem + (uint32_t)(wib * WAVE_LDS_BYTES);
    char* wave_buf = smem + wib * WAVE_LDS_BYTES;

    const int ntiles = (rows + TROWS - 1) / TROWS;

    int tile = gwave;
    if (tile < ntiles) stage_tile_async(H, rows, tile * TROWS, wave_lds, lane);  // prologue: buffer 0

    int pb = 0;                                // buffer holding the current tile
    for (; tile < ntiles; tile += nwave) {
        const int base = tile * TROWS;

        // Segment ids for this tile: lanes 0..15, one coalesced int64 load.
        int myseg = 0;
        {
            int irow = base + lane;
            if (lane < TROWS && irow < rows) myseg = (int)X[irow];
        }

        // Stage the NEXT tile into the other buffer, then wait until only its
        // 16 async ops can be outstanding -> the CURRENT tile's 16 are done.
        const int next = tile + nwave;
        if (next < ntiles) {
            stage_tile_async(H, rows, next * TROWS,
                             wave_lds + (uint32_t)((pb ^ 1) * TILE_BYTES), lane);
            asm volatile("s_wait_asynccnt 0x10" ::: "memory");
        } else {
            asm volatile("s_wait_asynccnt 0x0" ::: "memory");
        }

        // L2 prefetch two tiles ahead (gfx1250 global_prefetch_b8).
        const int pf = tile + 2 * nwave;
        if (pf < ntiles && lane < TROWS) {
            int pfrow = pf * TROWS + lane;
            if (pfrow < rows) __builtin_prefetch(H + (size_t)pfrow * COLS, 0, 1);
        }

        // Consume current tile from LDS (ds_load_b128) and scatter with
        // hardware f32 atomics (no return -> STOREcnt fire-and-forget).
        const char* buf = wave_buf + pb * TILE_BYTES;
        int cnt = rows - base;
        if (cnt > TROWS) cnt = TROWS;

        if (cnt == TROWS) {
            // Fast path: full tile, fully unrolled so the 16 ds_load_b128 and
            // 16 v_readlane_b32 issue ahead of the 64 atomic stores.
#pragma unroll
            for (int r = 0; r < TROWS; ++r) {
                int seg = __builtin_amdgcn_readlane(myseg, r);   // VALU broadcast
                const v4f v = *(const v4f*)(buf + r * ROW_BYTES + lane * 16);
                float* dst = out + (size_t)seg * COLS + (size_t)(lane * 4);
                unsafeAtomicAdd(dst + 0, v.x);
                unsafeAtomicAdd(dst + 1, v.y);
                unsafeAtomicAdd(dst + 2, v.z);
                unsafeAtomicAdd(dst + 3, v.w);
            }
        } else {
            for (int r = 0; r < cnt; ++r) {
                int seg = __builtin_amdgcn_readlane(myseg, r);   // uniform selector
                const v4f v = *(const v4f*)(buf + r * ROW_BYTES + lane * 16);
                float* dst = out + (size_t)seg * COLS + (size_t)(lane * 4);
                unsafeAtomicAdd(dst + 0, v.x);
                unsafeAtomicAdd(dst + 1, v.y);
                unsafeAtomicAdd(dst + 2, v.z);
                unsafeAtomicAdd(dst + 3, v.w);
            }
        }
        pb ^= 1;
    }
}

__global__ __launch_bounds__(256) void zero_kernel(float* __restrict__ out, int n4) {
    int i = blockIdx.x * blockDim.x + threadIdx.x;
    if (i < n4) {
        v4f z = {0.0f, 0.0f, 0.0f, 0.0f};
        ((v4f*)out)[i] = z;   // RT store: keep output lines resident in L2
    }
}

extern "C" void kernel_launch(void* const* d_in, const int* in_sizes, int n_in,
                              void* d_out, int out_size, void* d_ws, size_t ws_size,
                              hipStream_t stream) {
    const float*     H   = (const float*)d_in[0];      // 1.6M x 128 f32
    const long long* X   = (const long long*)d_in[1];  // 1.6M int64
    float*           out = (float*)d_out;              // 100K x 128 f32

    const int rows = in_sizes[1];                      // 1,600,000
    const int n4   = out_size / 4;                     // output float4 count

    // Phase 1: zero the output (harness poisons it with 0xAA).
    zero_kernel<<<(n4 + 255) / 256, 256, 0, stream>>>(out, n4);

    // Phase 2: async-staged scatter-add. 3200 blocks x 256 thr = 25,600 waves,
    // 100,000 16-row tiles -> ~4 tiles/wave; 128KB dynamic LDS per block
    // (2 blocks/WGP within the 320KB WGP LDS).
    seg_scatter<<<3200, 256, BLOCK_LDS_BYTES, stream>>>(H, X, out, rows);
}